// ThetaMemoryNetwork_46136538693998
// MI455X (gfx1250) — compile-verified
//
#include <hip/hip_runtime.h>
#include <stdint.h>

// ThetaMemoryNetwork (delta-rule fast weights), chunkwise-parallel exact
// reformulation for MI455X / gfx1250.  GEMMs on V_WMMA_F32_16X16X4_F32
// (f32 matches the fp32 reference numerics).  k_recur (the sequential
// critical path) double-buffers its per-chunk H / A staging with CDNA5
// async global->LDS copies (GLOBAL_LOAD_ASYNC_TO_LDS_B128 / ASYNCcnt),
// hiding the load latency behind the WMMA phases.
//
//   K1 : H[all tokens] = W_e @ x        (parallel, 512 blocks)
//   K1b: A[chunk]      = H^T H  (16x16) (parallel, 512 blocks)
//   K2 : per (batch, 16-row slab of theta), sequential over 64 chunks:
//          P0 = thetaSlab @ H (WMMA), tiny triangular err solve,
//          thetaSlab += LR * E H^T (WMMA), pred -> d_out
//        128 independent blocks (theta recurrence decomposes row-wise).
//   K3 : out = W_o @ pred, in-place in d_out (parallel, 512 blocks)
//
// Workspace: H = B*N*D floats (8 MB) + A_all = (B*N/16)*256 floats (512 KB).

typedef float v2f __attribute__((ext_vector_type(2)));
typedef float v8f __attribute__((ext_vector_type(8)));

#define D 256
#define C 16           // chunk length
#define LRF 0.01f
#define NTOK 8192      // B*N
#define NCHUNK 512     // NTOK/C
#define CPB 64         // chunks per batch (N/C)
#define HPAD 20        // Hc row stride (floats): 80B keeps 16B alignment for
                       // async B128 writes; stride-20 cols hit 16 distinct banks

// f32 WMMA 16x16x4:  D(16x16) = A(16x4) * B(4x16) + C
// A layout: lanes 0-15: v.x=K0, v.y=K1 ; lanes 16-31: v.x=K2, v.y=K3 (row = lane&15)
// B layout: lanes 0-15: v.x=K0, v.y=K1 ; lanes 16-31: v.x=K2, v.y=K3 (col = lane&15)
// C/D layout: vgpr r: lanes 0-15 -> (M=r, N=lane), lanes 16-31 -> (M=r+8, N=lane-16)
#define WMMA_F32(a, b, c) \
  __builtin_amdgcn_wmma_f32_16x16x4_f32(false, (a), false, (b), (short)0, (c), false, false)

// ---- CDNA5 async global->LDS copy (tracked by ASYNCcnt) ------------------
// LDS dest address comes from the vdst VGPR; low 32 bits of a generic LDS pointer are
// the wave-relative LDS byte offset (flat aperture truncation).
__device__ __forceinline__ void async_copy_b128(const float* gaddr, void* lds) {
  const uint32_t lds_off = (uint32_t)(uintptr_t)lds;
  asm volatile("global_load_async_to_lds_b128 %0, %1, off"
               :: "v"(lds_off), "v"(gaddr)
               : "memory");
}
__device__ __forceinline__ void wait_async0() {
  asm volatile("s_wait_asynccnt 0x0" ::: "memory");
}

// ---------------- Kernel 1: H = W_e @ X for all tokens --------------------
__global__ __launch_bounds__(256) void k_encode(const float* __restrict__ W_e,
                                                const float* __restrict__ seq,
                                                float* __restrict__ Hws) {
  __shared__ float Xc[16 * 257];            // [token][dim], padded
  const int cb  = blockIdx.x;               // chunk of 16 tokens
  const long g0 = (long)cb * C;
  const int tid = threadIdx.x;

  for (int c = 0; c < C; ++c)
    Xc[c * 257 + tid] = seq[(g0 + c) * D + tid];
  __syncthreads();

  const int wave = tid >> 5, lane = tid & 31;
  const int hf = lane >> 4, ml = lane & 15;

  for (int t = 0; t < 2; ++t) {             // 8 waves x 2 = 16 row tiles
    const int mBase = (wave + t * 8) * 16;
    v8f acc = {};
    for (int k0 = 0; k0 < D; k0 += 4) {
      v2f a, b;
      a.x = W_e[(mBase + ml) * D + k0 + 2 * hf];
      a.y = W_e[(mBase + ml) * D + k0 + 2 * hf + 1];
      b.x = Xc[ml * 257 + k0 + 2 * hf];       // B[k][c] = X[c][k]
      b.y = Xc[ml * 257 + k0 + 2 * hf + 1];
      acc = WMMA_F32(a, b, acc);
    }
    // H stored as [chunk][d][c] : ws[cb*4096 + d*16 + c]
    for (int r = 0; r < 8; ++r) {
      const int row = mBase + r + 8 * hf;
      Hws[(long)cb * (D * C) + row * C + ml] = acc[r];
    }
  }
}

// ---------------- Kernel 1b: A = H^T H per chunk --------------------------
__global__ __launch_bounds__(256) void k_hth(const float* __restrict__ Hws,
                                             float* __restrict__ Aws) {
  __shared__ float part[8 * 256];
  const int cb = blockIdx.x;
  const int tid = threadIdx.x, wave = tid >> 5, lane = tid & 31;
  const int hf = lane >> 4, ml = lane & 15;
  const long hb = (long)cb * (D * C);

  v8f acc = {};
  for (int k0 = wave * 32; k0 < wave * 32 + 32; k0 += 4) {
    // A[m][k] = H[k][m] and B[k][n] = H[k][n] -> identical per-lane values
    v2f h;
    h.x = Hws[hb + (k0 + 2 * hf) * C + ml];
    h.y = Hws[hb + (k0 + 2 * hf + 1) * C + ml];
    acc = WMMA_F32(h, h, acc);
  }
  for (int r = 0; r < 8; ++r)
    part[wave * 256 + (r + 8 * hf) * 16 + ml] = acc[r];
  __syncthreads();
  float s = 0.f;
  for (int w = 0; w < 8; ++w) s += part[w * 256 + tid];
  Aws[(long)cb * 256 + tid] = s;            // A[s][c] row-major
}

// ---------------- Kernel 2: theta recurrence per (batch, row-slab) --------
__global__ __launch_bounds__(64) void k_recur(const float* __restrict__ theta0,
                                              const float* __restrict__ tgt,
                                              const float* __restrict__ Hws,
                                              const float* __restrict__ Aws,
                                              float* __restrict__ pred_out) {
  __shared__ float thetaS[16 * 257];        // 16 theta rows, padded
  __shared__ float Hc[2][D * HPAD];         // H chunk [d][c], double-buffered
  __shared__ float Am[2][256];              // A = H^T H, double-buffered
  __shared__ float P0p[2 * 256];            // per-wave K-partials of P0
  __shared__ float P0[16 * 17];
  __shared__ float Eb[16 * 17];             // err slab [row][c]

  const int slab = blockIdx.x;              // 0..15 -> theta rows
  const int b    = blockIdx.y;              // 0..7 batch
  const int dBase = slab * 16;
  const int tid = threadIdx.x, wave = tid >> 5, lane = tid & 31;
  const int hf = lane >> 4, ml = lane & 15;

  // --- async prefetch of one chunk (H: 256 rows x 16 floats, A: 256 floats)
  auto prefetch_chunk = [&](int cb, int buf) {
    const float* hg = Hws + (long)cb * (D * C);
    for (int i = tid; i < 1024; i += 64) {       // 16B per lane-op
      const int r = i >> 2, q = (i & 3) * 4;
      async_copy_b128(hg + r * C + q, &Hc[buf][r * HPAD + q]);
    }
    async_copy_b128(Aws + (long)cb * 256 + tid * 4, &Am[buf][tid * 4]);
  };

  prefetch_chunk(b * CPB, 0);               // start chunk 0 DMA early

  for (int i = tid; i < 16 * D; i += 64) {
    const int r = i >> 8, c = i & 255;
    thetaS[r * 257 + c] = theta0[(dBase + r) * D + c];
  }
  __syncthreads();

  for (int ch = 0; ch < CPB; ++ch) {
    const int cb  = b * CPB + ch;
    const long g0 = (long)cb * C;
    const int buf = ch & 1;
    const float* HcC = Hc[buf];
    const float* AmC = Am[buf];

    wait_async0();                          // current buffer DMA complete
    __syncthreads();
    if (ch + 1 < CPB) prefetch_chunk(cb + 1, buf ^ 1);   // overlap with compute

    { // P0 = thetaS(16xD) @ H(DxC), K split over the 2 waves
      v8f acc = {};
      for (int k0 = wave * 128; k0 < wave * 128 + 128; k0 += 4) {
        v2f a, bb;
        a.x  = thetaS[ml * 257 + k0 + 2 * hf];
        a.y  = thetaS[ml * 257 + k0 + 2 * hf + 1];
        bb.x = HcC[(k0 + 2 * hf) * HPAD + ml];
        bb.y = HcC[(k0 + 2 * hf + 1) * HPAD + ml];
        acc = WMMA_F32(a, bb, acc);
      }
      for (int r = 0; r < 8; ++r)
        P0p[wave * 256 + (r + 8 * hf) * 16 + ml] = acc[r];
    }
    __syncthreads();
    for (int i = tid; i < 256; i += 64) {
      const int r = i >> 4, c = i & 15;
      P0[r * 17 + c] = P0p[i] + P0p[256 + i];
    }
    __syncthreads();

    // Sequential triangular err solve:
    //   err_c = tgt_c - P0[:,c] - LR * sum_{s<c} A[s][c] * err_s
    if (tid < 16) {
      const int d = dBase + tid;
      float e[16];
#pragma unroll
      for (int c = 0; c < C; ++c) {
        const float tv = tgt[(g0 + c) * D + d];
        float acc = tv - P0[tid * 17 + c];
#pragma unroll
        for (int s2 = 0; s2 < c; ++s2)
          acc -= LRF * AmC[s2 * 16 + c] * e[s2];
        e[c] = acc;
        Eb[tid * 17 + c] = acc;
        pred_out[(g0 + c) * D + d] = tv - acc;   // pred = tgt - err
      }
    }
    __syncthreads();

    // thetaS += LR * E(16x16) @ H^T(16xD): 16 n-tiles, K=16
    for (int nB = wave; nB < 16; nB += 2) {
      const int nBase = nB * 16;
      v8f acc = {};
      for (int k0 = 0; k0 < 16; k0 += 4) {
        v2f a, bb;
        a.x  = Eb[ml * 17 + k0 + 2 * hf];
        a.y  = Eb[ml * 17 + k0 + 2 * hf + 1];
        bb.x = HcC[(nBase + ml) * HPAD + k0 + 2 * hf];   // B[k][n] = H[n][k]
        bb.y = HcC[(nBase + ml) * HPAD + k0 + 2 * hf + 1];
        acc = WMMA_F32(a, bb, acc);
      }
      for (int r = 0; r < 8; ++r) {
        const int row = r + 8 * hf;
        thetaS[row * 257 + nBase + ml] += LRF * acc[r];
      }
    }
    __syncthreads();
  }
}

// ---------------- Kernel 3: out = W_o @ pred (in place in d_out) ----------
__global__ __launch_bounds__(256) void k_output(const float* __restrict__ W_o,
                                                float* __restrict__ io) {
  __shared__ float Pc[256 * 16];            // pred [dim][token]
  __shared__ float Oc[16 * 257];            // out  [token][dim], padded
  const int cb  = blockIdx.x;
  const long g0 = (long)cb * C;
  const int tid = threadIdx.x, wave = tid >> 5, lane = tid & 31;
  const int hf = lane >> 4, ml = lane & 15;

  for (int c = 0; c < C; ++c)
    Pc[tid * 16 + c] = io[(g0 + c) * D + tid];
  __syncthreads();

  for (int t = 0; t < 2; ++t) {
    const int mBase = (wave + t * 8) * 16;
    v8f acc = {};
    for (int k0 = 0; k0 < D; k0 += 4) {
      v2f a, bb;
      a.x  = W_o[(mBase + ml) * D + k0 + 2 * hf];
      a.y  = W_o[(mBase + ml) * D + k0 + 2 * hf + 1];
      bb.x = Pc[(k0 + 2 * hf) * 16 + ml];
      bb.y = Pc[(k0 + 2 * hf + 1) * 16 + ml];
      acc = WMMA_F32(a, bb, acc);
    }
    for (int r = 0; r < 8; ++r)
      Oc[ml * 257 + mBase + r + 8 * hf] = acc[r];
  }
  __syncthreads();
  for (int c = 0; c < C; ++c)
    io[(g0 + c) * D + tid] = Oc[c * 257 + tid];
}

extern "C" void kernel_launch(void* const* d_in, const int* in_sizes, int n_in,
                              void* d_out, int out_size, void* d_ws, size_t ws_size,
                              hipStream_t stream) {
  const float* seq    = (const float*)d_in[0];   // (8,1024,256)
  const float* tgt    = (const float*)d_in[1];   // (8,1024,256)
  const float* theta0 = (const float*)d_in[2];   // (256,256)
  const float* W_e    = (const float*)d_in[3];   // (256,256)
  const float* W_o    = (const float*)d_in[4];   // (256,256)
  float* out = (float*)d_out;                    // (8,1024,256)

  float* Hws = (float*)d_ws;                     // NTOK*D floats (8 MB)
  float* Aws = Hws + (size_t)NTOK * D;           // NCHUNK*256 floats (512 KB)

  k_encode<<<NCHUNK, 256, 0, stream>>>(W_e, seq, Hws);
  k_hth   <<<NCHUNK, 256, 0, stream>>>(Hws, Aws);
  dim3 g2(16, 8);                                // 16 row-slabs x 8 batches
  k_recur <<<g2, 64, 0, stream>>>(theta0, tgt, Hws, Aws, out);
  k_output<<<NCHUNK, 256, 0, stream>>>(W_o, out);
}